// MultiScale_84610855731504
// MI455X (gfx1250) — compile-verified
//
#include <hip/hip_runtime.h>
#include <hip/hip_bf16.h>

// ---------------------------------------------------------------------------
// Problem constants (from the reference)
// ---------------------------------------------------------------------------
#define N1  40000
#define N2  10000
#define KNN 16
#define C   64
#define UPC 128
#define S   8
#define CS  8
#define EPS 1e-5f

typedef __attribute__((ext_vector_type(2))) float v2f;
typedef __attribute__((ext_vector_type(8))) float v8f;

// ---------------------------------------------------------------------------
// WMMA fp32 GEMM:  Y[N,64] = X[N,KDIM] @ W[KDIM,64] + bias, optionally
// accumulating per-channel sum/sumsq for training-mode BatchNorm.
// Block = 256 threads (8 waves). Each block: 32 rows x 64 cols.
// Each wave: one 16x16 tile via V_WMMA_F32_16X16X4_F32 (K stepped by 4).
// X panel and full W staged in LDS (ds_* traffic, reused 4x / 2x).
// ---------------------------------------------------------------------------
template <int KDIM>
__global__ __launch_bounds__(256)
void gemm_bias_stats(const float* __restrict__ X, const float* __restrict__ W,
                     const float* __restrict__ bias, float* __restrict__ Y,
                     float* __restrict__ gsum, float* __restrict__ gsq, int N)
{
    __shared__ float lds_x[32 * KDIM];
    __shared__ float lds_w[KDIM * 64];

    const int tid = threadIdx.x;
    const int row0 = blockIdx.x * 32;

    for (int i = tid; i < KDIM * 64; i += 256) lds_w[i] = W[i];
    for (int i = tid; i < 32 * KDIM; i += 256) {
        int r = row0 + i / KDIM;
        lds_x[i] = (r < N) ? X[(size_t)r * KDIM + (i % KDIM)] : 0.0f;
    }
    __syncthreads();

    const int wave = tid >> 5;
    const int lane = tid & 31;
    const int mt = wave >> 2;          // 0..1  (16-row tile in block)
    const int nt = wave & 3;           // 0..3  (16-col tile)
    const int lo = lane & 15;
    const int hi = lane >> 4;

    v8f acc = {};
    const int arow = mt * 16 + lo;     // A: lanes 0-15 -> M, VGPR0/1 -> K pair
    const int bcol = nt * 16 + lo;     // B: row striped across lanes
    #pragma unroll
    for (int k0 = 0; k0 < KDIM; k0 += 4) {
        v2f a, b;
        a.x = lds_x[arow * KDIM + k0 + 2 * hi + 0];
        a.y = lds_x[arow * KDIM + k0 + 2 * hi + 1];
        b.x = lds_w[(k0 + 2 * hi + 0) * 64 + bcol];
        b.y = lds_w[(k0 + 2 * hi + 1) * 64 + bcol];
        acc = __builtin_amdgcn_wmma_f32_16x16x4_f32(
                  false, a, false, b, (short)0, acc, false, false);
    }

    const int col = nt * 16 + lo;
    const float bval = bias[col];
    const int rbase = row0 + mt * 16 + 8 * hi;   // C/D: VGPR i -> M = i (+8 hi)
    float psum = 0.f, psq = 0.f;
    #pragma unroll
    for (int i = 0; i < 8; ++i) {
        int r = rbase + i;
        float y = acc[i] + bval;
        if (r < N) {
            Y[(size_t)r * 64 + col] = y;
            psum += y; psq += y * y;
        }
    }
    if (gsum != nullptr) {
        psum += __shfl_xor(psum, 16, 32);   // lane l and l^16 share a column
        psq  += __shfl_xor(psq,  16, 32);
        if (lane < 16) {
            atomicAdd(&gsum[col], psum);
            atomicAdd(&gsq[col],  psq);
        }
    }
}

// stats -> per-channel affine (scale, shift) for y = x*scale + shift
__global__ void bn_finalize(const float* __restrict__ sum, const float* __restrict__ sq,
                            const float* __restrict__ gamma, const float* __restrict__ beta,
                            float inv_cnt, int nch,
                            float* __restrict__ scale, float* __restrict__ shift)
{
    int c = blockIdx.x * blockDim.x + threadIdx.x;
    if (c < nch) {
        float mean = sum[c] * inv_cnt;
        float var  = sq[c] * inv_cnt - mean * mean;
        float s    = rsqrtf(var + EPS) * gamma[c];
        scale[c] = s;
        shift[c] = beta[c] - mean * s;
    }
}

__global__ void bn_relu_apply(const float* __restrict__ Y,
                              const float* __restrict__ scale,
                              const float* __restrict__ shift,
                              float* __restrict__ out, int total)
{
    int i = blockIdx.x * blockDim.x + threadIdx.x;
    if (i < total) {
        int c = i & 63;
        float v = fmaf(Y[i], scale[c], shift[c]);
        out[i] = v > 0.f ? v : 0.f;
    }
}

// 3-NN inverse-distance interpolation: x2i[n,c] = sum_j iw_j * x2h[idx_j, c]
__global__ void interp_kernel(const float* __restrict__ p1, const float* __restrict__ p2,
                              const int* __restrict__ iidx, const float* __restrict__ x2h,
                              float* __restrict__ x2i)
{
    int g = blockIdx.x * blockDim.x + threadIdx.x;
    int n = g >> 6, c = g & 63;
    if (n >= N1) return;
    int i0 = iidx[n * 3 + 0], i1 = iidx[n * 3 + 1], i2 = iidx[n * 3 + 2];
    float px = p1[n * 3 + 0], py = p1[n * 3 + 1], pz = p1[n * 3 + 2];
    float w[3]; int id[3] = {i0, i1, i2};
    float tot = 0.f;
    #pragma unroll
    for (int j = 0; j < 3; ++j) {
        float dx = px - p2[id[j] * 3 + 0];
        float dy = py - p2[id[j] * 3 + 1];
        float dz = pz - p2[id[j] * 3 + 2];
        float d  = sqrtf(dx * dx + dy * dy + dz * dz);
        w[j] = 1.0f / (d + 1e-8f);
        tot += w[j];
    }
    float inv = 1.0f / tot;
    x2i[(size_t)n * 64 + c] = inv * (w[0] * x2h[(size_t)i0 * 64 + c] +
                                     w[1] * x2h[(size_t)i1 * 64 + c] +
                                     w[2] * x2h[(size_t)i2 * 64 + c]);
}

// per-channel sum/sumsq of w[n,k,c] = xk[knn[n,k],c] - xq[n,c] over all (n,k)
__global__ __launch_bounds__(256)
void w_stats(const float* __restrict__ xk, const float* __restrict__ xq,
             const int* __restrict__ knn, float* __restrict__ gsum,
             float* __restrict__ gsq)
{
    const int tid = threadIdx.x;
    const int c = tid & 63, grp = tid >> 6;           // 4 groups of 64
    const int P = N1 * KNN;
    float s = 0.f, q = 0.f;
    for (int p = blockIdx.x * 4 + grp; p < P; p += gridDim.x * 4) {
        int n = p >> 4;
        int idx = knn[p];                              // knn flat index == p
        __builtin_prefetch(&xk[(size_t)idx * 64 + c], 0, 1);
        float v = xk[(size_t)idx * 64 + c] - xq[(size_t)n * 64 + c];
        s += v; q += v * v;
    }
    __shared__ float shs[256], shq[256];
    shs[tid] = s; shq[tid] = q;
    __syncthreads();
    if (tid < 64) {
        s = shs[tid] + shs[tid + 64] + shs[tid + 128] + shs[tid + 192];
        q = shq[tid] + shq[tid + 64] + shq[tid + 128] + shq[tid + 192];
        atomicAdd(&gsum[tid], s);
        atomicAdd(&gsq[tid],  q);
    }
}

// t[p,0:8] = relu(bn0(xk[knn[p]] - xq[p>>4])) @ Ww1 + bw1 ; accumulate BN1 stats
__global__ __launch_bounds__(256)
void weight_mlp1(const float* __restrict__ xk, const float* __restrict__ xq,
                 const int* __restrict__ knn,
                 const float* __restrict__ sc0, const float* __restrict__ sh0,
                 const float* __restrict__ Ww1, const float* __restrict__ bw1,
                 float* __restrict__ t, float* __restrict__ gsum, float* __restrict__ gsq)
{
    const int P = N1 * KNN;
    int p = blockIdx.x * blockDim.x + threadIdx.x;
    float a[CS];
    #pragma unroll
    for (int j = 0; j < CS; ++j) a[j] = 0.f;
    if (p < P) {
        int n = p >> 4;
        int idx = knn[p];
        #pragma unroll
        for (int j = 0; j < CS; ++j) a[j] = bw1[j];
        for (int c = 0; c < 64; ++c) {
            float v = xk[(size_t)idx * 64 + c] - xq[(size_t)n * 64 + c];
            v = fmaf(v, sc0[c], sh0[c]);
            v = v > 0.f ? v : 0.f;
            #pragma unroll
            for (int j = 0; j < CS; ++j) a[j] = fmaf(v, Ww1[c * CS + j], a[j]);
        }
        #pragma unroll
        for (int j = 0; j < CS; ++j) t[(size_t)p * CS + j] = a[j];
    }
    // wave-level reduction of BN1 stats (inactive threads contribute zeros)
    #pragma unroll
    for (int j = 0; j < CS; ++j) {
        float s = (p < P) ? a[j] : 0.f;
        float q = s * s;
        #pragma unroll
        for (int off = 16; off >= 1; off >>= 1) {
            s += __shfl_xor(s, off, 32);
            q += __shfl_xor(q, off, 32);
        }
        if ((threadIdx.x & 31) == 0) {
            atomicAdd(&gsum[j], s);
            atomicAdd(&gsq[j],  q);
        }
    }
}

// per point: bn1+relu -> 8x8 matmul -> softmax over K -> shared-plane agg + residual
__global__ __launch_bounds__(64)
void weight_mlp2_agg(const float* __restrict__ t,
                     const float* __restrict__ sc1, const float* __restrict__ sh1,
                     const float* __restrict__ Ww2, const float* __restrict__ bw2,
                     const float* __restrict__ xv, const int* __restrict__ knn,
                     const float* __restrict__ x1h, float* __restrict__ out)
{
    const int n = blockIdx.x;
    const int tid = threadIdx.x;           // 64 threads
    __shared__ float wsm[KNN][CS];
    __shared__ int   sidx[KNN];
    if (tid < KNN) sidx[tid] = knn[n * KNN + tid];
    // 16*8 = 128 logits with 64 threads (2 each)
    #pragma unroll
    for (int p = tid; p < KNN * CS; p += 64) {
        int k = p >> 3, j = p & 7;
        float accv = bw2[j];
        #pragma unroll
        for (int c = 0; c < CS; ++c) {
            float v = fmaf(t[(size_t)(n * KNN + k) * CS + c], sc1[c], sh1[c]);
            v = v > 0.f ? v : 0.f;
            accv = fmaf(v, Ww2[c * CS + j], accv);
        }
        wsm[k][j] = accv;
    }
    __syncthreads();
    if (tid < CS) {                        // softmax over neighbors, per plane j
        int j = tid;
        float mx = -3.402823466e38f;
        #pragma unroll
        for (int k = 0; k < KNN; ++k) mx = fmaxf(mx, wsm[k][j]);
        float ssum = 0.f;
        #pragma unroll
        for (int k = 0; k < KNN; ++k) { float e = __expf(wsm[k][j] - mx); wsm[k][j] = e; ssum += e; }
        float inv = 1.0f / ssum;
        #pragma unroll
        for (int k = 0; k < KNN; ++k) wsm[k][j] *= inv;
    }
    __syncthreads();
    const int c = tid, j = c & 7;
    float accv = x1h[(size_t)n * 64 + c];
    #pragma unroll
    for (int k = 0; k < KNN; ++k)
        accv = fmaf(xv[(size_t)sidx[k] * 64 + c], wsm[k][j], accv);
    out[(size_t)n * 64 + c] = accv;
}

// ---------------------------------------------------------------------------
// Orchestration
// ---------------------------------------------------------------------------
extern "C" void kernel_launch(void* const* d_in, const int* in_sizes, int n_in,
                              void* d_out, int out_size, void* d_ws, size_t ws_size,
                              hipStream_t stream)
{
    (void)in_sizes; (void)n_in; (void)out_size; (void)ws_size;
    const float* p1   = (const float*)d_in[0];
    const float* x1   = (const float*)d_in[1];
    const float* p2   = (const float*)d_in[2];
    const float* x2   = (const float*)d_in[3];
    const int*   knn  = (const int*)  d_in[4];
    const int*   iidx = (const int*)  d_in[5];
    const float* W1   = (const float*)d_in[6];
    const float* b1   = (const float*)d_in[7];
    const float* g1   = (const float*)d_in[8];
    const float* be1  = (const float*)d_in[9];
    const float* W2   = (const float*)d_in[10];
    const float* b2   = (const float*)d_in[11];
    const float* g2   = (const float*)d_in[12];
    const float* be2  = (const float*)d_in[13];
    const float* Wq   = (const float*)d_in[14];
    const float* bq   = (const float*)d_in[15];
    const float* Wk   = (const float*)d_in[16];
    const float* bk   = (const float*)d_in[17];
    const float* Wv   = (const float*)d_in[18];
    const float* bvv  = (const float*)d_in[19];
    const float* gw0  = (const float*)d_in[20];
    const float* bw0  = (const float*)d_in[21];
    const float* Ww1  = (const float*)d_in[22];
    const float* bw1  = (const float*)d_in[23];
    const float* gw1  = (const float*)d_in[24];
    const float* bew1 = (const float*)d_in[25];
    const float* Ww2  = (const float*)d_in[26];
    const float* bw2  = (const float*)d_in[27];
    float* out = (float*)d_out;

    // workspace layout (floats)
    float* ws = (float*)d_ws;
    float* y1pre = ws;                         // N1*64
    float* y2pre = y1pre + (size_t)N1 * 64;    // N2*64
    float* x1h   = y2pre + (size_t)N2 * 64;    // N1*64
    float* x2h   = x1h   + (size_t)N1 * 64;    // N2*64
    float* x2i   = x2h   + (size_t)N2 * 64;    // N1*64
    float* xq    = x2i   + (size_t)N1 * 64;    // N1*64
    float* xk    = xq    + (size_t)N1 * 64;    // N1*64
    float* xvb   = xk    + (size_t)N1 * 64;    // N1*64
    float* tbuf  = xvb   + (size_t)N1 * 64;    // N1*KNN*CS
    float* stats = tbuf  + (size_t)N1 * KNN * CS;
    float* s1sum = stats;            float* s1sq  = s1sum + 64;
    float* s2sum = s1sq  + 64;       float* s2sq  = s2sum + 64;
    float* w0sum = s2sq  + 64;       float* w0sq  = w0sum + 64;
    float* w1sum = w0sq  + 64;       float* w1sq  = w1sum + CS;
    float* bn1s  = w1sq  + CS;       float* bn1h  = bn1s  + 64;
    float* bn2s  = bn1h  + 64;       float* bn2h  = bn2s  + 64;
    float* bw0s  = bn2h  + 64;       float* bw0h  = bw0s  + 64;
    float* bw1s  = bw0h  + 64;       float* bw1h  = bw1s  + CS;
    const size_t stats_bytes = (size_t)((bw1h + CS) - stats) * sizeof(float);

    hipMemsetAsync(stats, 0, stats_bytes, stream);

    // linear1 / linear2 pre-activations + BN stats
    gemm_bias_stats<64> <<<(N1 + 31) / 32, 256, 0, stream>>>(x1, W1, b1, y1pre, s1sum, s1sq, N1);
    gemm_bias_stats<128><<<(N2 + 31) / 32, 256, 0, stream>>>(x2, W2, b2, y2pre, s2sum, s2sq, N2);
    bn_finalize<<<1, 64, 0, stream>>>(s1sum, s1sq, g1, be1, 1.0f / (float)N1, 64, bn1s, bn1h);
    bn_finalize<<<1, 64, 0, stream>>>(s2sum, s2sq, g2, be2, 1.0f / (float)N2, 64, bn2s, bn2h);
    bn_relu_apply<<<(N1 * 64 + 255) / 256, 256, 0, stream>>>(y1pre, bn1s, bn1h, x1h, N1 * 64);
    bn_relu_apply<<<(N2 * 64 + 255) / 256, 256, 0, stream>>>(y2pre, bn2s, bn2h, x2h, N2 * 64);

    // 3-NN interpolation to p1
    interp_kernel<<<(N1 * 64 + 255) / 256, 256, 0, stream>>>(p1, p2, iidx, x2h, x2i);

    // q/k/v projections (WMMA, no stats)
    gemm_bias_stats<64><<<(N1 + 31) / 32, 256, 0, stream>>>(x1h, Wq, bq,  xq,  nullptr, nullptr, N1);
    gemm_bias_stats<64><<<(N1 + 31) / 32, 256, 0, stream>>>(x2i, Wk, bk,  xk,  nullptr, nullptr, N1);
    gemm_bias_stats<64><<<(N1 + 31) / 32, 256, 0, stream>>>(x2i, Wv, bvv, xvb, nullptr, nullptr, N1);

    // BN0 stats over w = x_k - x_q  (N1*K elements per channel)
    w_stats<<<1024, 256, 0, stream>>>(xk, xq, knn, w0sum, w0sq);
    bn_finalize<<<1, 64, 0, stream>>>(w0sum, w0sq, gw0, bw0, 1.0f / (float)(N1 * KNN), 64, bw0s, bw0h);

    // weight-MLP layer 1 (+ BN1 stats)
    weight_mlp1<<<(N1 * KNN + 255) / 256, 256, 0, stream>>>(xk, xq, knn, bw0s, bw0h,
                                                            Ww1, bw1, tbuf, w1sum, w1sq);
    bn_finalize<<<1, CS, 0, stream>>>(w1sum, w1sq, gw1, bew1, 1.0f / (float)(N1 * KNN), CS, bw1s, bw1h);

    // weight-MLP layer 2 + softmax + shared-plane aggregation + residual
    weight_mlp2_agg<<<N1, 64, 0, stream>>>(tbuf, bw1s, bw1h, Ww2, bw2, xvb, knn, x1h, out);
}